// SpikeToCalcium_68015102099996
// MI455X (gfx1250) — compile-verified
//
#include <hip/hip_runtime.h>

typedef __attribute__((ext_vector_type(2))) float v2f;
typedef __attribute__((ext_vector_type(4))) float v4f;
typedef __attribute__((ext_vector_type(8))) float v8f;

#define B_TOTAL 1000
#define T_OUT   20000
#define KT      181
#define U_W     (T_OUT + KT - 1)            // 20180 columns in u (mult of 4)
#define BT      16                          // batch tile (N dim)
#define WAVES   8
#define TILES_PER_WAVE 4
#define T_BLOCK (WAVES * TILES_PER_WAVE * 16)   // 512 t-outputs per block
#define NEED    (T_BLOCK + KT - 1)          // 692 u columns staged per row (= 4*173)
#define STRIDE  700                         // LDS row stride: 700 mod 64 = 60 = 4*15
                                            // -> conflict-free b64 fragment reads
#define WPAD    216                         // 16 front pad + 181 taps + back pad
#define NP      49                          // ceil((181+15)/4) K-chunks of 4

__global__ __launch_bounds__(256)
void fir_wmma_f32(const float* __restrict__ u, const float* __restrict__ ker,
                  float* __restrict__ out) {
  __shared__ float wlds[WPAD];
  __shared__ float ulds[BT * STRIDE];

  const int tid = threadIdx.x;
  const int b0  = blockIdx.y * BT;
  const int t0  = blockIdx.x * T_BLOCK;

  // Stage reversed + zero-padded kernel: wlds[16 + j] = w[j] = ker[KT-1-j]
  if (tid < WPAD) {
    float v = 0.0f;
    int j = tid - 16;
    if (j >= 0 && j < KT) v = ker[KT - 1 - j];
    wlds[tid] = v;
  }

  // Stage u tile: BT rows x NEED floats, as float4 (rows/cols clamped; clamped
  // data only feeds outputs that are never stored).
  const int row_quads = NEED / 4;           // 173
  for (int i = tid; i < BT * row_quads; i += 256) {
    int br = i / row_quads;
    int c4 = i - br * row_quads;
    int gb = b0 + br; if (gb >= B_TOTAL) gb = B_TOTAL - 1;
    long gcol = (long)t0 + 4 * c4;
    if (gcol > (long)(U_W - 4)) gcol = U_W - 4;
    v4f val = *(const v4f*)(u + (long)gb * U_W + gcol);
    *(v4f*)(&ulds[br * STRIDE + 4 * c4]) = val;
  }
  __syncthreads();

  const int lane = tid & 31;
  const int wave = tid >> 5;
  const int half = lane >> 4;               // 0 -> K=0,1 ; 1 -> K=2,3 (A/B layout)
  const int nb   = lane & 15;               // A-row M / B-col N (= batch)
  const int k0   = half * 2;
  const int wt   = wave * (TILES_PER_WAVE * 16);  // wave's local t offset

  v8f acc0 = {}, acc1 = {}, acc2 = {}, acc3 = {};

  // A_p[m,k] = w[4p + k - m] -> wlds[16 + 4p + k0 - nb (+1)]
  const float* wrow = &wlds[16 + k0 - nb];
  // B_p[k,n] = u_lds[n][wt + 16*tile + 4p + k]  (8B aligned: base mult of 4 + even)
  const float* urow = &ulds[nb * STRIDE + wt + k0];

  #pragma unroll 7
  for (int p = 0; p < NP; ++p) {
    v2f a;
    a.x = wrow[4 * p];
    a.y = wrow[4 * p + 1];
    v2f f0 = *(const v2f*)(urow + 4 * p);
    v2f f1 = *(const v2f*)(urow + 4 * p + 16);
    v2f f2 = *(const v2f*)(urow + 4 * p + 32);
    v2f f3 = *(const v2f*)(urow + 4 * p + 48);
    acc0 = __builtin_amdgcn_wmma_f32_16x16x4_f32(false, a, false, f0, (short)0, acc0, false, false);
    acc1 = __builtin_amdgcn_wmma_f32_16x16x4_f32(false, a, false, f1, (short)0, acc1, false, false);
    acc2 = __builtin_amdgcn_wmma_f32_16x16x4_f32(false, a, false, f2, (short)0, acc2, false, false);
    acc3 = __builtin_amdgcn_wmma_f32_16x16x4_f32(false, a, false, f3, (short)0, acc3, false, false);
  }

  // Store: lane holds N = nb (batch), VGPR r holds M = r + 8*half (t offset).
  // Consecutive M -> consecutive t -> two 16B stores per tile per lane.
  const bool bok  = (b0 + nb) < B_TOTAL;
  const long orow = (long)(b0 + nb) * T_OUT;

  v8f accs[TILES_PER_WAVE] = {acc0, acc1, acc2, acc3};
  #pragma unroll
  for (int j = 0; j < TILES_PER_WAVE; ++j) {
    int tb = t0 + wt + 16 * j;              // tile base t (multiple of 16)
    if (bok && tb < T_OUT) {                // T_OUT % 16 == 0: no straddle
      v8f a = accs[j];
      v4f lo = __builtin_shufflevector(a, a, 0, 1, 2, 3);
      v4f hi = __builtin_shufflevector(a, a, 4, 5, 6, 7);
      float* dst = out + orow + tb + 8 * half;
      *(v4f*)(dst)     = lo;
      *(v4f*)(dst + 4) = hi;
    }
  }
}

extern "C" void kernel_launch(void* const* d_in, const int* in_sizes, int n_in,
                              void* d_out, int out_size, void* d_ws, size_t ws_size,
                              hipStream_t stream) {
  const float* u   = (const float*)d_in[0];
  const float* ker = (const float*)d_in[1];
  float* out = (float*)d_out;

  dim3 grid((T_OUT + T_BLOCK - 1) / T_BLOCK,   // 40
            (B_TOTAL + BT - 1) / BT);          // 63
  fir_wmma_f32<<<grid, 256, 0, stream>>>(u, ker, out);
}